// S4DKernel_40596030882436
// MI455X (gfx1250) — compile-verified
//
#include <hip/hip_runtime.h>

typedef float v2f __attribute__((ext_vector_type(2)));
typedef float v8f __attribute__((ext_vector_type(8)));

#define NH 32        // N/2 state pairs
#define TL 64        // tile edge: L = TL*TL = 4096
#define STRIDE 68    // padded LDS row stride (floats)

__global__ __launch_bounds__(512) void s4d_wmma_kernel(
    const float* __restrict__ log_dt,
    const float* __restrict__ C,           // (H, NH, 2)
    const float* __restrict__ log_A_real,  // (H, NH)
    const float* __restrict__ A_imag,      // (H, NH)
    float* __restrict__ out)               // (H, L)
{
    __shared__ float cDtAr[NH], cDtAi[NH], cWr[NH], cWi[NH];
    __shared__ float lA[TL * STRIDE];   // A[q][k] : k=2n -> Re(W z^{64q}), k=2n+1 -> -Im(...)
    __shared__ float lBt[TL * STRIDE];  // Bt[r][k]: k=2n -> Re(z^r),       k=2n+1 ->  Im(z^r)

    const int h   = blockIdx.x;
    const int tid = threadIdx.x;

    // ---- Phase 0: per-n constants: dtA = A*dt, W = 2 * C * (exp(dtA)-1)/A ----
    if (tid < NH) {
        const int n = tid;
        float dt = __expf(log_dt[h]);
        float Ar = -__expf(log_A_real[h * NH + n]);
        float Ai = A_imag[h * NH + n];
        float Cr = C[(h * NH + n) * 2 + 0];
        float Ci = C[(h * NH + n) * 2 + 1];
        float dtAr = Ar * dt, dtAi = Ai * dt;
        float e  = __expf(dtAr);
        float er = e * __cosf(dtAi) - 1.0f;     // exp(dtA) - 1
        float ei = e * __sinf(dtAi);
        float inv = 1.0f / (Ar * Ar + Ai * Ai); // divide by A
        float qr = (er * Ar + ei * Ai) * inv;
        float qi = (ei * Ar - er * Ai) * inv;
        cDtAr[n] = dtAr;
        cDtAi[n] = dtAi;
        cWr[n] = 2.0f * (Cr * qr - Ci * qi);    // fold the final *2 into W
        cWi[n] = 2.0f * (Cr * qi + Ci * qr);
    }
    __syncthreads();

    // ---- Phase 1: build 64x64 real tiles A (rows q) and Bt (rows r) ----
    // out[q,r] = sum_n Re( (W z^{64q})_n * (z^r)_n ),  K-expansion: re,-im | re,im
    #pragma unroll
    for (int i = 0; i < (NH * TL) / 512; ++i) {
        int idx = tid + i * 512;
        int n = idx >> 6;     // 0..31
        int t = idx & 63;     // 0..63
        float dtAr = cDtAr[n], dtAi = cDtAi[n];

        // Bt: z^t  (t = r), stored K-contiguous per row
        float ft  = (float)t;
        float mag = __expf(dtAr * ft);
        lBt[t * STRIDE + 2 * n]     = mag * __cosf(dtAi * ft);
        lBt[t * STRIDE + 2 * n + 1] = mag * __sinf(dtAi * ft);

        // A: W * z^{64 t}  (t = q), K-contiguous per row
        float fs   = 64.0f * ft;
        float mag2 = __expf(dtAr * fs);
        float Zr = mag2 * __cosf(dtAi * fs);
        float Zi = mag2 * __sinf(dtAi * fs);
        float Wr = cWr[n], Wi = cWi[n];
        lA[t * STRIDE + 2 * n]     =  Wr * Zr - Wi * Zi;   //  Re(W z)
        lA[t * STRIDE + 2 * n + 1] = -(Wr * Zi + Wi * Zr); // -Im(W z)
    }
    __syncthreads();

    // ---- Phase 2: 64x64x64 f32 GEMM via V_WMMA_F32_16X16X4_F32 ----
    // 16 waves, one 16x16 output tile each.
    const int wave = tid >> 5;
    const int lane = tid & 31;
    const int hi   = lane >> 4;    // half-wave select
    const int l15  = lane & 15;
    const int m16  = wave >> 2;    // output tile row  (q block)
    const int n16  = wave & 3;     // output tile col  (r block)

    // A operand (16x4 f32): lanes 0-15 hold K=k0,k0+1 (k0=4kk); lanes 16-31 K=k0+2,k0+3; M = lane&15
    // B operand (4x16 f32): same K split per half-wave; N = lane&15  -> contiguous in Bt rows
    const float* Arow = &lA [(m16 * 16 + l15) * STRIDE];
    const float* Brow = &lBt[(n16 * 16 + l15) * STRIDE];

    v8f acc = {};
    #pragma unroll
    for (int kk = 0; kk < 16; ++kk) {
        int k0 = 4 * kk + 2 * hi;
        v2f a, b;
        a.x = Arow[k0];
        a.y = Arow[k0 + 1];
        b.x = Brow[k0];
        b.y = Brow[k0 + 1];
        // D = A(16x4) * B(4x16) + C
        acc = __builtin_amdgcn_wmma_f32_16x16x4_f32(
            /*neg_a=*/false, a, /*neg_b=*/false, b,
            /*c_mod=*/(short)0, acc, /*reuse_a=*/false, /*reuse_b=*/false);
    }

    // ---- Store: C/D layout: VGPR j -> M=j (lanes 0-15) / M=j+8 (lanes 16-31)
    float* orow = out + (size_t)h * (TL * TL);
    const int col = n16 * 16 + l15;
    #pragma unroll
    for (int j = 0; j < 8; ++j) {
        int m = m16 * 16 + j + 8 * hi;
        orow[m * TL + col] = acc[j];
    }
}

extern "C" void kernel_launch(void* const* d_in, const int* in_sizes, int n_in,
                              void* d_out, int out_size, void* d_ws, size_t ws_size,
                              hipStream_t stream) {
    const float* log_dt     = (const float*)d_in[0];
    const float* C          = (const float*)d_in[1];
    const float* log_A_real = (const float*)d_in[2];
    const float* A_imag     = (const float*)d_in[3];
    // d_in[4] is L (==4096, structural: tiling assumes L = 64*64)
    float* out = (float*)d_out;

    const int H = in_sizes[0];
    s4d_wmma_kernel<<<H, 512, 0, stream>>>(log_dt, C, log_A_real, A_imag, out);
}